// time_embeddings_566935683729
// MI455X (gfx1250) — compile-verified
//
#include <hip/hip_runtime.h>
#include <hip/hip_bf16.h>

// time_embeddings: out[b,i] = sin(ang) for even i, cos(ang) for odd i,
// ang = time[b] * 10000^(-2*(i/2)/1280).  B=65536, DIM=1280, f32.
//
// HBM-bound (335.5 MB write stream -> ~14.4 us floor @ 23.3 TB/s).
// 1 thread = 4 consecutive columns (2 sin/cos pairs); 320 threads/block
// (10 wave32) cover a full row; each block walks 32 rows so the per-column
// rates (2x v_exp_f32/thread) amortize. cos folded into sin via +0.25-turn
// phase (v_sin_f32 takes turns) -> no divergence, exactly 1 FMA + 1 v_sin_f32
// per element. time[row] is block-uniform -> scalar s_load. Output leaves via
// contiguous non-temporal global_store_b128 (don't rinse the 192 MB L2 with a
// 335 MB one-shot stream). Full blocks take a branch-free unrolled path.

typedef __attribute__((ext_vector_type(4))) float v4f;  // native vector for nontemporal builtin

#define TE_DIM  1280
#define TE_COLG (TE_DIM / 4)      // 320 threads per block, 10 wave32
#define TE_ROWS 32                // rows per block

__global__ __launch_bounds__(TE_COLG) void
time_embeddings_kernel(const float* __restrict__ time_in,
                       float* __restrict__ out,
                       int batch)
{
    const int cg   = threadIdx.x;                 // column group: columns 4cg..4cg+3
    const int row0 = blockIdx.x * TE_ROWS;

    // pair index p = i >> 1 ; for this thread p in {2cg, 2cg+1}
    // rate_s(p) = 10000^(-2p/1280) / (2*pi)
    //           = exp2( p * (-2*log2(1e4)/1280) - log2(2*pi) )
    const float k = -0.02076205062866211f;        // -2*log2(10000)/1280
    const float b = -2.6514961294723187f;         // -log2(2*pi)
    const float p0 = (float)(2 * cg);
    const float e0 = __builtin_amdgcn_exp2f(fmaf(p0,        k, b)); // rate for i=4cg+0,1
    const float e1 = __builtin_amdgcn_exp2f(fmaf(p0 + 1.0f, k, b)); // rate for i=4cg+2,3

    v4f* __restrict__ orow = reinterpret_cast<v4f*>(out) + cg;

    if (row0 + TE_ROWS <= batch) {
        // Hot path: full block, branch-free body. Unroll 8 so consecutive
        // block-uniform time loads can merge into wider s_load_bNN.
#pragma unroll 8
        for (int r = 0; r < TE_ROWS; ++r) {
            const int row = row0 + r;
            const float t = time_in[row];         // block-uniform -> scalar load
            v4f v;
            v.x = __builtin_amdgcn_sinf(t * e0);              // even col -> sin
            v.y = __builtin_amdgcn_sinf(fmaf(t, e0, 0.25f));  // odd  col -> cos(x)=sin(x+1/4 turn)
            v.z = __builtin_amdgcn_sinf(t * e1);
            v.w = __builtin_amdgcn_sinf(fmaf(t, e1, 0.25f));
            __builtin_nontemporal_store(v, orow + (size_t)row * (TE_DIM / 4));
        }
    } else {
        // Tail path: partial last block only.
        for (int r = 0; r < TE_ROWS; ++r) {
            const int row = row0 + r;
            if (row < batch) {
                const float t = time_in[row];
                v4f v;
                v.x = __builtin_amdgcn_sinf(t * e0);
                v.y = __builtin_amdgcn_sinf(fmaf(t, e0, 0.25f));
                v.z = __builtin_amdgcn_sinf(t * e1);
                v.w = __builtin_amdgcn_sinf(fmaf(t, e1, 0.25f));
                __builtin_nontemporal_store(v, orow + (size_t)row * (TE_DIM / 4));
            }
        }
    }
}

extern "C" void kernel_launch(void* const* d_in, const int* in_sizes, int n_in,
                              void* d_out, int out_size, void* d_ws, size_t ws_size,
                              hipStream_t stream)
{
    (void)n_in; (void)out_size; (void)d_ws; (void)ws_size;
    const float* time_in = (const float*)d_in[0];
    float*       out     = (float*)d_out;
    const int    batch   = in_sizes[0];           // 65536

    const int blocks = (batch + TE_ROWS - 1) / TE_ROWS;   // 2048 blocks x 10 wave32
    time_embeddings_kernel<<<blocks, TE_COLG, 0, stream>>>(time_in, out, batch);
}